// FCEloss_alpha_25031069401300
// MI455X (gfx1250) — compile-verified
//
#include <hip/hip_runtime.h>
#include <hip/hip_bf16.h>

// FCE loss: fused masked-softmax loss over pred[B,100], B=524288.
// Memory-bound streaming reduction (210 MB @ 23.3 TB/s ~ 9us). Uses CDNA5
// async global->LDS DMA (GLOBAL_LOAD_ASYNC_TO_LDS_B128 / ASYNCcnt) with
// per-wave double buffering; no matrix math in the op, so no WMMA.

#define C_CLASSES 100
#define WAVES_PER_BLOCK 8
#define ROWS_PER_TILE 4            // 4 rows * 400B = 1600B = 100 x b128
#define TILE_BYTES (ROWS_PER_TILE * C_CLASSES * 4)

__device__ __forceinline__ void issue_tile_async(const float* __restrict__ pred,
                                                 long long tile,
                                                 float* ldsbuf, int lane) {
    const char* g = (const char*)pred + tile * (long long)TILE_BYTES;
    unsigned ldsBase = (unsigned)(unsigned long long)ldsbuf;  // low 32b = LDS offset
#pragma unroll
    for (int i = 0; i < 4; ++i) {
        int idx = i * 32 + lane;                 // b128 element index, 100 valid
        if (idx < C_CLASSES) {
            unsigned long long ga = (unsigned long long)(g + idx * 16);
            unsigned la = ldsBase + (unsigned)(idx * 16);
            // CDNA5 async DMA: memory -> LDS, tracked by ASYNCcnt
            asm volatile("global_load_async_to_lds_b128 %0, %1, off"
                         :: "v"(la), "v"(ga)
                         : "memory");
        }
    }
}

__global__ void __launch_bounds__(256)
fce_main_kernel(const float* __restrict__ pred,
                const int*  __restrict__ teacher,
                const float* __restrict__ per_cls_w,
                float* __restrict__ partial,
                int numTiles) {
    __shared__ __align__(16) float smem[WAVES_PER_BLOCK][2][ROWS_PER_TILE * C_CLASSES];
    __shared__ float sAcc[WAVES_PER_BLOCK];

    const int lane = threadIdx.x & 31;
    const int wave = threadIdx.x >> 5;
    const long long stride = (long long)gridDim.x * WAVES_PER_BLOCK;
    long long t = (long long)blockIdx.x * WAVES_PER_BLOCK + wave;

    float acc = 0.0f;
    int buf = 0;

    if (t < numTiles) issue_tile_async(pred, t, &smem[wave][0][0], lane);

    for (; t < numTiles; t += stride, buf ^= 1) {
        const long long tn = t + stride;
        if (tn < numTiles) {
            // WAR fence: DMA must not overwrite LDS the wave is still reading.
            asm volatile("s_wait_dscnt 0" ::: "memory");
            issue_tile_async(pred, tn, &smem[wave][buf ^ 1][0], lane);
            asm volatile("s_wait_asynccnt 4" ::: "memory");  // current tile done
        } else {
            asm volatile("s_wait_asynccnt 0" ::: "memory");
        }

        const float* base = &smem[wave][buf][0];
#pragma unroll
        for (int r = 0; r < ROWS_PER_TILE; ++r) {
            const float* row = base + r * C_CLASSES;
            float v0 = row[lane];
            float v1 = row[lane + 32];
            float v2 = row[lane + 64];
            float v3 = (lane < 4) ? row[lane + 96] : -3.0e38f;

            // wave32 butterfly max
            float m = fmaxf(fmaxf(v0, v1), fmaxf(v2, v3));
#pragma unroll
            for (int off = 16; off; off >>= 1)
                m = fmaxf(m, __shfl_xor(m, off, 32));

            float e0 = __expf(v0 - m);
            float e1 = __expf(v1 - m);
            float e2 = __expf(v2 - m);
            float e3 = __expf(v3 - m);

            const int tr = teacher[t * ROWS_PER_TILE + r];  // broadcast load

            float sm = 0.0f;
            if (v0 > 0.0f && (lane)      != tr) sm += e0;
            if (v1 > 0.0f && (lane + 32) != tr) sm += e1;
            if (v2 > 0.0f && (lane + 64) != tr) sm += e2;
            if (v3 > 0.0f && (lane + 96) != tr) sm += e3;
            float sa = e0 + e1 + e2 + e3;

#pragma unroll
            for (int off = 16; off; off >>= 1) {
                sm += __shfl_xor(sm, off, 32);
                sa += __shfl_xor(sa, off, 32);
            }

            const float w  = per_cls_w[tr];
            const float yk = w * sm / sa;
            acc += __logf(yk + 1e-7f);
        }
    }

    if (lane == 0) sAcc[wave] = acc;
    __syncthreads();
    if (threadIdx.x == 0) {
        float s = 0.0f;
#pragma unroll
        for (int i = 0; i < WAVES_PER_BLOCK; ++i) s += sAcc[i];
        partial[blockIdx.x] = s;
    }
}

__global__ void __launch_bounds__(256)
fce_reduce_kernel(const float* __restrict__ partial, float* __restrict__ out,
                  int n, float invB) {
    __shared__ float s[256];
    float a = 0.0f;
    for (int i = threadIdx.x; i < n; i += 256) a += partial[i];
    s[threadIdx.x] = a;
    __syncthreads();
#pragma unroll
    for (int off = 128; off > 0; off >>= 1) {
        if ((int)threadIdx.x < off) s[threadIdx.x] += s[threadIdx.x + off];
        __syncthreads();
    }
    if (threadIdx.x == 0) out[0] = s[0] * invB;
}

extern "C" void kernel_launch(void* const* d_in, const int* in_sizes, int n_in,
                              void* d_out, int out_size, void* d_ws, size_t ws_size,
                              hipStream_t stream) {
    (void)n_in; (void)out_size;
    const float* pred      = (const float*)d_in[0];
    const int*   teacher   = (const int*)d_in[1];
    const float* per_cls_w = (const float*)d_in[2];
    float* out = (float*)d_out;
    float* ws  = (float*)d_ws;

    const int B = in_sizes[1];                 // 524288 rows
    const int numTiles = B / ROWS_PER_TILE;    // B divisible by 4

    int blocks = 1024;
    if (ws_size < (size_t)blocks * sizeof(float)) {
        blocks = (int)(ws_size / sizeof(float));
        if (blocks < 1) blocks = 1;
    }

    fce_main_kernel<<<blocks, 256, 0, stream>>>(pred, teacher, per_cls_w, ws, numTiles);
    fce_reduce_kernel<<<1, 256, 0, stream>>>(ws, out, blocks, 1.0f / (float)B);
}